// Encoder_32229434589360
// MI455X (gfx1250) — compile-verified
//
#include <hip/hip_runtime.h>
#include <hip/hip_bf16.h>
#include <math.h>

// Fragment/vector types for CDNA5 wave32 WMMA
typedef __attribute__((ext_vector_type(2))) float v2f;  // A/B frag of V_WMMA_F32_16X16X4_F32
typedef __attribute__((ext_vector_type(8))) float v8f;  // C/D frag (16x16 f32)

#define NCH1 128
#define NCH2 64

// ---------------- degree / normalization ----------------

__global__ __launch_bounds__(256) void init_deg_kernel(float* __restrict__ deg, int n) {
  int i = blockIdx.x * 256 + threadIdx.x;
  if (i < n) deg[i] = 1.0f;  // self-loop contributes +1
}

__global__ __launch_bounds__(256) void degree_kernel(const int* __restrict__ dst,
                                                     float* __restrict__ deg, int ne) {
  int e = blockIdx.x * 256 + threadIdx.x;
  if (e < ne) atomicAdd(&deg[dst[e]], 1.0f);
}

__global__ __launch_bounds__(256) void norm_kernel(const float* __restrict__ deg,
                                                   float* __restrict__ dinv,
                                                   float* __restrict__ snorm, int n) {
  int i = blockIdx.x * 256 + threadIdx.x;
  if (i < n) {
    float d = deg[i];
    dinv[i]  = rsqrtf(d);
    snorm[i] = 1.0f / d;
  }
}

// ---------------- fp32 WMMA GEMM: out[N,COUT] = act(H)[N,K] @ W[K,COUT] ----------------
// One wave computes a 16x16 tile; (COUT/16) waves per block cover 16 rows x COUT cols.
// W is staged through LDS in KC-row chunks (cooperative float4 loads).
// RELU_IN fuses the previous layer's bias+ReLU into the A-fragment load
// (saves a full read+write pass over the intermediate activation).

template<int K, int COUT, bool RELU_IN>
__global__ __launch_bounds__((COUT / 16) * 32) void gemm_wmma_kernel(
    const float* __restrict__ H, const float* __restrict__ W,
    const float* __restrict__ bin, float* __restrict__ out) {
  constexpr int KC = 64;                 // K-chunk staged in LDS (KC*COUT*4 bytes)
  constexpr int NT = (COUT / 16) * 32;   // threads per block
  __shared__ float Wlds[KC * COUT];

  const int tid    = threadIdx.x;
  const int wave   = tid >> 5;
  const int lane   = tid & 31;
  const int lane16 = lane & 15;
  const int hi     = (lane >> 4) & 1;    // lanes 16-31 hold K+2/K+3 (A) rows / M+8 (C)
  const int row0   = blockIdx.x * 16;    // N is a multiple of 16 (50000 = 3125*16)
  const int col0   = wave * 16;

  v8f c = {};
  for (int kb = 0; kb < K; kb += KC) {
    __syncthreads();
    // cooperative stage of W[kb:kb+KC, :] into LDS
    const float4* gw = (const float4*)(W + (size_t)kb * COUT);
    float4*       lw = (float4*)Wlds;
    for (int i = tid; i < KC * COUT / 4; i += NT) lw[i] = gw[i];
    __syncthreads();

    // A-frag source: row (row0+lane16), columns kb+k0+{0,1} (lo lanes) / +{2,3} (hi lanes)
    const float* arow  = H + (size_t)(row0 + lane16) * K + kb + 2 * hi;
    const float* brow  = RELU_IN ? (bin + kb + 2 * hi) : nullptr;
    // B-frag source from LDS: rows k0+{0,1}/{2,3}, column col0+lane16
    const float* bbase = Wlds + 2 * hi * COUT + col0 + lane16;

#pragma unroll
    for (int k0 = 0; k0 < KC; k0 += 4) {
      v2f a, b;
      float ax = arow[k0];
      float ay = arow[k0 + 1];
      if constexpr (RELU_IN) {  // fused previous-layer bias + ReLU (bias is L1-resident)
        ax = fmaxf(ax + brow[k0], 0.0f);
        ay = fmaxf(ay + brow[k0 + 1], 0.0f);
      }
      a.x = ax; a.y = ay;
      b.x = bbase[(size_t)k0 * COUT];
      b.y = bbase[(size_t)(k0 + 1) * COUT];
      // D = A(16x4) * B(4x16) + C, exact fp32
      c = __builtin_amdgcn_wmma_f32_16x16x4_f32(
          /*neg_a=*/false, a, /*neg_b=*/false, b,
          /*c_mod=*/(short)0, c, /*reuse_a=*/false, /*reuse_b=*/false);
    }
  }

  // C layout: VGPR r -> M = r (lanes 0-15) / M = r+8 (lanes 16-31), N = lane16
  float* orow = out + (size_t)(row0 + 8 * hi) * COUT + col0 + lane16;
#pragma unroll
  for (int r = 0; r < 8; ++r) orow[(size_t)r * COUT] = c[r];
}

// ---------------- self-loop init: agg = hw * snorm[node] ----------------

template<int C>
__global__ __launch_bounds__(256) void self_init_kernel(const float* __restrict__ hw,
                                                        const float* __restrict__ snorm,
                                                        float* __restrict__ agg, int total) {
  int idx = blockIdx.x * 256 + threadIdx.x;
  if (idx < total) agg[idx] = hw[idx] * snorm[idx / C];
}

// ---------------- edge scatter: one wave32 per edge, VEC channels per lane ----------------
// Edge id is forced wave-uniform (readfirstlane) so src/dst/dinv loads become
// scalar (SMEM) loads instead of 32x-redundant vector loads; the feature gather
// is an explicit b128/b64 with SGPR base + lane offset. Scatter is dword
// global_atomic_add_f32 into the L2-resident accumulator.

template<int C, int VEC>
__global__ __launch_bounds__(256) void scatter_kernel(
    const float* __restrict__ hw, const int* __restrict__ src, const int* __restrict__ dst,
    const float* __restrict__ dinv, float* __restrict__ agg, int ne) {
  static_assert(32 * VEC == C, "wave covers all channels");
  const int lane = threadIdx.x & 31;
  const int e = __builtin_amdgcn_readfirstlane(blockIdx.x * 8 + (threadIdx.x >> 5));
  if (e >= ne) return;                 // uniform exit
  const int s = src[e];                // scalar loads (wave-uniform index)
  const int d = dst[e];
  const float w = dinv[s] * dinv[d];   // symmetric GCN edge norm, wave-uniform
  const float* hs = hw + (size_t)s * C + lane * VEC;
  float*       ad = agg + (size_t)d * C + lane * VEC;
  if constexpr (VEC == 4) {
    float4 t = *(const float4*)hs;     // global_load_b128
    atomicAdd(&ad[0], t.x * w);
    atomicAdd(&ad[1], t.y * w);
    atomicAdd(&ad[2], t.z * w);
    atomicAdd(&ad[3], t.w * w);
  } else {
    float2 t = *(const float2*)hs;     // global_load_b64
    atomicAdd(&ad[0], t.x * w);
    atomicAdd(&ad[1], t.y * w);
  }
}

// ---------------- fused bias + ReLU (in place, final layer epilogue) ----------------

template<int C>
__global__ __launch_bounds__(256) void bias_relu_kernel(float* __restrict__ agg,
                                                        const float* __restrict__ b, int total) {
  int idx = blockIdx.x * 256 + threadIdx.x;
  if (idx < total) {
    float v = agg[idx] + b[idx & (C - 1)];
    agg[idx] = v > 0.0f ? v : 0.0f;
  }
}

// ---------------- launch ----------------

extern "C" void kernel_launch(void* const* d_in, const int* in_sizes, int n_in,
                              void* d_out, int out_size, void* d_ws, size_t ws_size,
                              hipStream_t stream) {
  const float* x  = (const float*)d_in[0];
  const int*   ei = (const int*)d_in[1];
  const float* W1 = (const float*)d_in[2];
  const float* b1 = (const float*)d_in[3];
  const float* W2 = (const float*)d_in[4];
  const float* b2 = (const float*)d_in[5];

  const int n  = in_sizes[0] / NCH1;  // 50000 (multiple of 16)
  const int ne = in_sizes[1] / 2;     // 800000
  const int* src = ei;
  const int* dst = ei + ne;

  // workspace carve-out (floats): deg | dinv | snorm | hw1 | agg1 | hw2  (~65 MB)
  float* ws    = (float*)d_ws;
  float* deg   = ws;  ws += n;
  float* dinv  = ws;  ws += n;
  float* snorm = ws;  ws += n;
  float* hw1   = ws;  ws += (size_t)n * NCH1;
  float* agg1  = ws;  ws += (size_t)n * NCH1;  // pre-activation; ReLU fused into GEMM2
  float* hw2   = ws;  ws += (size_t)n * NCH2;
  float* agg2  = (float*)d_out;       // accumulate layer-2 output directly in d_out

  const int nbN   = (n + 255) / 256;
  const int nbE   = (ne + 255) / 256;
  const int nbE8  = (ne + 7) / 8;     // scatter: 8 waves (edges) per 256-thread block
  const int nb1   = (int)(((size_t)n * NCH1 + 255) / 256);
  const int nb2   = (int)(((size_t)n * NCH2 + 255) / 256);

  // normalization terms
  init_deg_kernel<<<nbN, 256, 0, stream>>>(deg, n);
  degree_kernel<<<nbE, 256, 0, stream>>>(dst, deg, ne);
  norm_kernel<<<nbN, 256, 0, stream>>>(deg, dinv, snorm, n);

  // ---- layer 1: agg1 = scatter(x@W1) + (x@W1)*snorm   (bias+ReLU deferred into GEMM2) ----
  gemm_wmma_kernel<NCH1, NCH1, false><<<n / 16, 256, 0, stream>>>(x, W1, nullptr, hw1);
  self_init_kernel<NCH1><<<nb1, 256, 0, stream>>>(hw1, snorm, agg1, n * NCH1);
  scatter_kernel<NCH1, 4><<<nbE8, 256, 0, stream>>>(hw1, src, dst, dinv, agg1, ne);

  // ---- layer 2: hw2 = relu(agg1 + b1) @ W2 ; out = relu(scatter(hw2) + hw2*snorm + b2) ----
  gemm_wmma_kernel<NCH1, NCH2, true><<<n / 16, 128, 0, stream>>>(agg1, W2, b1, hw2);
  self_init_kernel<NCH2><<<nb2, 256, 0, stream>>>(hw2, snorm, agg2, n * NCH2);
  scatter_kernel<NCH2, 2><<<nbE8, 256, 0, stream>>>(hw2, src, dst, dinv, agg2, ne);
  bias_relu_kernel<NCH2><<<nb2, 256, 0, stream>>>(agg2, b2, n * NCH2);
}